// WordTaggingParserModel_53420803228468
// MI455X (gfx1250) — compile-verified
//
#include <hip/hip_runtime.h>
#include <hip/hip_bf16.h>

typedef __attribute__((ext_vector_type(2))) float v2f;
typedef __attribute__((ext_vector_type(8))) float v8f;

#define B_    16
#define S_    4096
#define L_    2048
#define DLM   768
#define DPOS  64
#define DH    832          // 768 + 64
#define NCLS  50
#define NPAD  64           // padded class dim (4 x 16 n-tiles)
#define HSTRIDE 836        // 832 + 4 pad -> stride % 64 == 4, bank-conflict free
#define KHALF 416          // 832 / 2, multiple of 4

__device__ __forceinline__ int lower_bound_i(const int* __restrict__ a, int n, int v) {
    int lo = 0, hi = n;
    while (lo < hi) {
        int mid = (lo + hi) >> 1;
        if (a[mid] < v) lo = mid + 1; else hi = mid;
    }
    return lo;
}

// ---- prep: WT[n][k] = (n < 50) ? W[k][n] : 0   (zero-padded transpose) ------
__global__ __launch_bounds__(256)
void transpose_pad_w_kernel(const float* __restrict__ Wm,   // [DH, NCLS]
                            float*       __restrict__ WT)   // [NPAD, DH]
{
    const int idx = blockIdx.x * 256 + threadIdx.x;   // over NPAD*DH
    if (idx >= NPAD * DH) return;
    const int n = idx / DH;
    const int k = idx - n * DH;
    WT[idx] = (n < NCLS) ? Wm[k * NCLS + n] : 0.0f;
}

__global__ __launch_bounds__(256)
void wordtag_fused_kernel(const int*   __restrict__ words,     // [B,S]
                          const int*   __restrict__ word_ids,  // [B,S] sorted per row
                          const int*   __restrict__ pos,       // [B,L]
                          const float* __restrict__ E_lm,      // [VOCAB, DLM]
                          const float* __restrict__ E_pos,     // [N_POS, DPOS]
                          const float* __restrict__ WT,        // [NPAD, DH] transposed+padded
                          const float* __restrict__ bias,      // [NCLS]
                          float*       __restrict__ out)       // [B, L, NCLS]
{
    __shared__ float hTile[16 * HSTRIDE];     // 53,504 B: 16 x 832 h-tile
    __shared__ float partBuf[8 * 256];        //  8,192 B: per-wave WMMA partials

    const int tid  = threadIdx.x;
    const int lane = tid & 31;
    const int wv   = tid >> 5;                // wave id 0..7 (wave32)
    const int b    = blockIdx.x >> 7;         // / (L/16 = 128)
    const int l0   = (blockIdx.x & 127) << 4; // first of 16 word rows

    // ---------------- Stage 1: build 16 x 832 tile of h in LDS ----------------
    const int* wid_row = word_ids + (size_t)b * S_;
    #pragma unroll
    for (int rr = 0; rr < 2; ++rr) {
        const int r = wv * 2 + rr;           // row within tile, 0..15
        const int l = l0 + r;
        const int seg_lo = lower_bound_i(wid_row, S_, l);
        const int seg_hi = lower_bound_i(wid_row, S_, l + 1);
        // segment-max of gathered E_lm rows, clamped at 0 (init = 0 handles
        // both the ReLU-clamp and empty segments, matching the reference)
        for (int d = lane; d < DLM; d += 32) {
            float m = 0.0f;
            for (int s = seg_lo; s < seg_hi; ++s) {
                const int w = words[(size_t)b * S_ + s];   // uniform scalar load
                m = fmaxf(m, E_lm[(size_t)w * DLM + d]);   // coalesced per-lane
            }
            hTile[r * HSTRIDE + d] = m;
        }
        // positional embedding slice (padding_idx row is zero in the table)
        const int p = pos[(size_t)b * L_ + l];
        for (int d = lane; d < DPOS; d += 32)
            hTile[r * HSTRIDE + DLM + d] = E_pos[p * DPOS + d];
    }
    __syncthreads();

    // ---------------- Stage 2: 16x64 GEMM tile via V_WMMA_F32_16X16X4_F32 -----
    // 8 waves: n-tile = wv&3, K-half = wv>>2. WMMA runs on every wave
    // unconditionally (EXEC all-1s requirement); zero-padded WT makes every
    // lane's column valid -> branch-free hot loop.
    const int ml    = lane & 15;      // M row (A) / N col (B,D)
    const int hl    = lane >> 4;      // lane-half selector
    const int kbase = (wv >> 2) * KHALF;
    const int ncol  = ((wv & 3) << 4) + ml;          // 0..63, in-bounds of WT
    const float* __restrict__ wcol = WT + (size_t)ncol * DH;

    v8f acc = {};
    for (int kk = 0; kk < KHALF; kk += 4) {
        const int k0 = kbase + kk;
        // A 16x4 f32: lanes 0-15 -> K={k0,k0+1}, lanes 16-31 -> K={k0+2,k0+3}
        const v2f a  = *(const v2f*)(&hTile[ml * HSTRIDE + k0 + 2 * hl]);
        // B 4x16 f32 mirrors A; contiguous 8B load from transposed W
        const v2f bf = *(const v2f*)(wcol + k0 + 2 * hl);
        acc = __builtin_amdgcn_wmma_f32_16x16x4_f32(
                  false, a, false, bf, (short)0, acc, false, false);
    }

    // spill per-wave partial C to LDS
    #pragma unroll
    for (int r8 = 0; r8 < 8; ++r8)
        partBuf[wv * 256 + r8 * 32 + lane] = acc[r8];
    __syncthreads();

    // combine the two K-halves, add bias, write the 50 valid columns
    for (int idx = tid; idx < 4 * 256; idx += 256) {
        const float v = partBuf[idx] + partBuf[idx + 4 * 256];
        const int ln  = idx & 31;
        const int r8  = (idx >> 5) & 7;
        const int wvn = idx >> 8;                 // n-tile id 0..3
        const int n   = (wvn << 4) + (ln & 15);
        const int m   = r8 + 8 * (ln >> 4);
        if (n < NCLS)
            out[((size_t)(b * L_ + l0 + m)) * NCLS + n] = v + bias[n];
    }
}

extern "C" void kernel_launch(void* const* d_in, const int* in_sizes, int n_in,
                              void* d_out, int out_size, void* d_ws, size_t ws_size,
                              hipStream_t stream) {
    (void)in_sizes; (void)n_in; (void)out_size; (void)ws_size;
    const int*   words    = (const int*)  d_in[0];
    const int*   word_ids = (const int*)  d_in[1];
    const int*   pos      = (const int*)  d_in[2];
    const float* E_lm     = (const float*)d_in[3];
    const float* E_pos    = (const float*)d_in[4];
    const float* Wm       = (const float*)d_in[5];
    const float* bias     = (const float*)d_in[6];
    float*       out      = (float*)d_out;
    float*       WT       = (float*)d_ws;          // NPAD*DH floats = 208 KB

    const int prep_elems = NPAD * DH;
    transpose_pad_w_kernel<<<(prep_elems + 255) / 256, 256, 0, stream>>>(Wm, WT);

    const int grid = B_ * (L_ / 16);   // 2048 workgroups
    wordtag_fused_kernel<<<grid, 256, 0, stream>>>(
        words, word_ids, pos, E_lm, E_pos, WT, bias, out);
}